// loss_7825430413956
// MI455X (gfx1250) — compile-verified
//
#include <hip/hip_runtime.h>
#include <hip/hip_bf16.h>

typedef float v2f __attribute__((ext_vector_type(2)));
typedef float v8f __attribute__((ext_vector_type(8)));

// Single-instruction xor-lane exchange on wave32 via ds_swizzle_b32:
// group-of-32 mode, and_mask=0x1f, or_mask=0, xor_mask=k -> offset=(k<<10)|0x1f
template <int K>
__device__ __forceinline__ float xor_add(float v) {
    int s = __builtin_amdgcn_ds_swizzle(__float_as_int(v), (K << 10) | 0x1f);
    return v + __int_as_float(s);
}

// ---------------------------------------------------------------------------
// Wave reduction via V_WMMA_F32_16X16X4_F32 (full f32 precision):
//   A = 16x4 all-ones, B = 4x16 holding the 32 lane values (plus zeros).
//   D[m][n] = sum_k B[k][n] -> 16 column sums; 4 ds_swizzle steps finish.
// Must be called with EXEC == all ones (uniform call, blockDim % 32 == 0).
// ---------------------------------------------------------------------------
__device__ __forceinline__ float wave_reduce_wmma(float v) {
    v2f a = {1.0f, 1.0f};   // every lane contributes {1,1} -> A all-ones
    v2f b = {v, 0.0f};      // 32 values + 32 zeros fill the 4x16 B matrix
    v8f c = {};
    c = __builtin_amdgcn_wmma_f32_16x16x4_f32(
        false, a, false, b, (short)0, c, false, false);
    float w = c[0];          // lanes 0..15: D[0][n]; lanes 16..31: D[8][n] (same)
    w = xor_add<1>(w);
    w = xor_add<2>(w);
    w = xor_add<4>(w);
    w = xor_add<8>(w);       // lanes 0..15 now hold the full 32-lane sum
    return w;
}

// Block reduction (blockDim.x == 256 -> 8 waves). Result valid on thread 0.
__device__ __forceinline__ float block_reduce(float v) {
    __shared__ float red[8];
    __syncthreads();                 // safe reuse across successive calls
    float w = wave_reduce_wmma(v);
    int lane = threadIdx.x & 31;
    int wave = threadIdx.x >> 5;
    if (lane == 0) red[wave] = w;
    __syncthreads();
    float tot = 0.0f;
    if (threadIdx.x == 0) {
        #pragma unroll
        for (int i = 0; i < 8; ++i) tot += red[i];
    }
    return tot;
}

// BCE-with-logits, numerically stable (native v_exp_f32 / v_log_f32).
__device__ __forceinline__ float bce1(float x, float y) {
    float ax = __builtin_fabsf(x);
    float e  = __expf(-ax);
    float l  = __logf(1.0f + e);
    return fmaxf(x, 0.0f) - x * y + l;
}

// ---------------------------------------------------------------------------
// Kernel 1: streaming BCE partial sums. 16 outstanding b128 loads per thread
// before any math -> deep LOADcnt pipeline for HBM saturation.
// ---------------------------------------------------------------------------
#define BCE_ITERS 8

__global__ void bce_partial(const float4* __restrict__ x,
                            const float4* __restrict__ y,
                            float* __restrict__ partials, int n4) {
    const int stride = gridDim.x * blockDim.x;
    const int base   = blockIdx.x * blockDim.x + threadIdx.x;

    float4 a[BCE_ITERS], b[BCE_ITERS];
    #pragma unroll
    for (int k = 0; k < BCE_ITERS; ++k) {
        int i = base + k * stride;
        if (i < n4) { a[k] = x[i]; b[k] = y[i]; }
        else        { a[k] = make_float4(0,0,0,0); b[k] = make_float4(0,0,0,0); }
    }

    float s = 0.0f;
    #pragma unroll
    for (int k = 0; k < BCE_ITERS; ++k) {
        int i = base + k * stride;
        if (i < n4) {
            s += bce1(a[k].x, b[k].x);
            s += bce1(a[k].y, b[k].y);
            s += bce1(a[k].z, b[k].z);
            s += bce1(a[k].w, b[k].w);
        }
    }
    // Tail safety if grid*ITERS doesn't cover n4 (not the case at 2048x256x8).
    for (int i = base + BCE_ITERS * stride; i < n4; i += stride) {
        float4 aa = x[i], bb = y[i];
        s += bce1(aa.x, bb.x) + bce1(aa.y, bb.y) + bce1(aa.z, bb.z) + bce1(aa.w, bb.w);
    }

    float tot = block_reduce(s);
    if (threadIdx.x == 0) partials[blockIdx.x] = tot;
}

// ---------------------------------------------------------------------------
// Kernel 2: tiny losses (l1, l3) + l2 finalize. Single block, deterministic.
// ---------------------------------------------------------------------------
__global__ void finalize(const float* __restrict__ cls,
                         const int* __restrict__ label,
                         const float* __restrict__ newe,
                         const float* __restrict__ gte,
                         const float* __restrict__ partials, int nPartials,
                         float* __restrict__ out, int nPts, int bs) {
    int tid = threadIdx.x;

    // l1: cross-entropy over [bs, 2] logits
    float s1 = 0.0f;
    for (int b = tid; b < bs; b += blockDim.x) {
        float x0 = cls[2 * b], x1 = cls[2 * b + 1];
        float m  = fmaxf(x0, x1);
        float lse = m + __logf(__expf(x0 - m) + __expf(x1 - m));
        float xl = (label[b] == 0) ? x0 : x1;
        s1 += (lse - xl);
    }

    // l3: smooth-L1 on point-wise euclidean distances
    float s3 = 0.0f;
    for (int i = tid; i < nPts; i += blockDim.x) {
        float dx = gte[2 * i]     - newe[2 * i];
        float dy = gte[2 * i + 1] - newe[2 * i + 1];
        float d  = __builtin_sqrtf(dx * dx + dy * dy + 1e-10f);
        s3 += (d < 1.0f) ? 0.5f * d * d : d - 0.5f;
    }

    // l2: sum per-block partials
    float s2 = 0.0f;
    for (int i = tid; i < nPartials; i += blockDim.x) s2 += partials[i];

    float t1 = block_reduce(s1);
    float t2 = block_reduce(s2);
    float t3 = block_reduce(s3);

    if (tid == 0) {
        float inv = 1.0f / (float)bs;
        out[0] = t1 * inv;   // l1 (mean over bs)
        out[1] = t2 * inv;   // l2 (sum / bs)
        out[2] = t3 * inv;   // l3 (sum / bs)
    }
}

extern "C" void kernel_launch(void* const* d_in, const int* in_sizes, int n_in,
                              void* d_out, int out_size, void* d_ws, size_t ws_size,
                              hipStream_t stream) {
    const float* cls  = (const float*)d_in[0];   // [64, 2]
    const int*   lab  = (const int*)d_in[1];     // [64]
    const float* newe = (const float*)d_in[2];   // [64, 60, 2]
    const float* gte  = (const float*)d_in[3];   // [64, 60, 2]
    const float* ef   = (const float*)d_in[4];   // [64, 1, 512, 512]
    const float* ie   = (const float*)d_in[5];   // [64, 1, 512, 512]

    int nBce = in_sizes[4];          // 16,777,216
    int n4   = nBce / 4;             // float4 count = 4,194,304
    int nPts = in_sizes[2] / 2;      // 3840 points
    int bs   = in_sizes[1];          // 64

    float* partials = (float*)d_ws;  // 2048 floats = 8 KB scratch
    const int NBLK = 2048;           // 2048*256*8 float4s == n4 exactly

    bce_partial<<<NBLK, 256, 0, stream>>>((const float4*)ef, (const float4*)ie,
                                          partials, n4);
    finalize<<<1, 256, 0, stream>>>(cls, lab, newe, gte, partials, NBLK,
                                    (float*)d_out, nPts, bs);
}